// HarmonicOscillator_7335804141990
// MI455X (gfx1250) — compile-verified
//
#include <hip/hip_runtime.h>

// Problem sizes (fixed by the reference).
#define BB 4
#define TT 131072
#define HH 128
#define CH 1024                    // t-values per main block / scan chunk
#define NCHUNK ((BB * TT) / CH)    // 512 chunks total
#define CPB (TT / CH)              // 128 chunks per batch
#define PIPE 8                     // async global->LDS pipeline depth per wave

#if defined(__gfx1250__) && __has_builtin(__builtin_amdgcn_global_load_async_to_lds_b128) && \
    __has_builtin(__builtin_amdgcn_s_wait_asynccnt)
#define HAVE_ASYNC_LDS 1
typedef int v4i __attribute__((ext_vector_type(4)));
typedef __attribute__((address_space(1))) v4i* v4i_global;
typedef __attribute__((address_space(3))) v4i* v4i_lds;
#endif

// fract(x) = x - floor(x) as a single v_fract_f32 when available.
__device__ __forceinline__ float fracf(float x) {
#if __has_builtin(__builtin_amdgcn_fractf)
    return __builtin_amdgcn_fractf(x);
#else
    return x - floorf(x);
#endif
}

// ---------------------------------------------------------------------------
// Wave32 full butterfly sum. DPP path: quad_perm xor1/xor2, row_ror:4/8
// (rotation is a valid fold once quads hold uniform partials), then
// v_permlanex16 with identity selects for the lane^16 exchange.
// ---------------------------------------------------------------------------
#if __has_builtin(__builtin_amdgcn_update_dpp)
template <int CTRL>
__device__ __forceinline__ float dpp_movf(float x) {
    return __builtin_bit_cast(
        float, __builtin_amdgcn_update_dpp(0, __builtin_bit_cast(int, x), CTRL, 0xF, 0xF, true));
}
#endif

__device__ __forceinline__ float wave32_sum(float x) {
#if __has_builtin(__builtin_amdgcn_update_dpp)
    x += dpp_movf<0xB1>(x);   // quad_perm [1,0,3,2]  : lane ^ 1
    x += dpp_movf<0x4E>(x);   // quad_perm [2,3,0,1]  : lane ^ 2
    x += dpp_movf<0x124>(x);  // row_ror:4            : fold quads across 16-row
    x += dpp_movf<0x128>(x);  // row_ror:8            : fold octs across 16-row
#if __has_builtin(__builtin_amdgcn_permlanex16)
    x += __builtin_bit_cast(
        float, __builtin_amdgcn_permlanex16(__builtin_bit_cast(int, x), __builtin_bit_cast(int, x),
                                            0x76543210, 0xFEDCBA98, false, false));
#elif __has_builtin(__builtin_amdgcn_ds_swizzle)
    x += __builtin_bit_cast(
        float, __builtin_amdgcn_ds_swizzle(__builtin_bit_cast(int, x), 0x401F));  // xor 16
#else
    x += __shfl_xor(x, 16, 32);
#endif
    return x;
#else
#pragma unroll
    for (int d = 16; d > 0; d >>= 1) x += __shfl_xor(x, d, 32);
    return x;
#endif
}

// ---------------------------------------------------------------------------
// K1: per-chunk sums of up_phase (chunk = 1024 floats, block = 256 threads)
// ---------------------------------------------------------------------------
__global__ __launch_bounds__(256) void k_chunk_sums(const float* __restrict__ up,
                                                    float* __restrict__ chunk_sums) {
    const int tid  = threadIdx.x;
    const int lane = tid & 31;
    const int wid  = tid >> 5;
    const long base = (long)blockIdx.x * CH + (long)tid * 4;
    const float4 u = *(const float4*)(up + base);
    float v = wave32_sum((u.x + u.y) + (u.z + u.w));
    __shared__ float wsum[8];
    if (lane == 0) wsum[wid] = v;
    __syncthreads();
    if (tid == 0) {
        float s = 0.f;
#pragma unroll
        for (int w = 0; w < 8; ++w) s += wsum[w];
        chunk_sums[blockIdx.x] = s;
    }
}

// ---------------------------------------------------------------------------
// K2: segmented (per-batch, 128-chunk segments) EXCLUSIVE scan of chunk sums.
// One block, NCHUNK(=512) threads, Hillis-Steele in LDS.
// ---------------------------------------------------------------------------
__global__ __launch_bounds__(NCHUNK) void k_scan_chunks(const float* __restrict__ chunk_sums,
                                                        float* __restrict__ chunk_prefix) {
    __shared__ float buf[NCHUNK];
    const int tid = threadIdx.x;
    const int c   = tid & (CPB - 1);      // chunk index within its batch segment
    const float s = chunk_sums[tid];
    buf[tid] = s;
    __syncthreads();
    for (int off = 1; off < CPB; off <<= 1) {
        float v = buf[tid];
        float n = (c >= off) ? buf[tid - off] : 0.0f;
        __syncthreads();
        buf[tid] = v + n;
        __syncthreads();
    }
    chunk_prefix[tid] = buf[tid] - s;     // inclusive -> exclusive
}

// ---------------------------------------------------------------------------
// K3: main pass. Block = 256 threads = 8 wave32s, owns CH=1024 t-values.
//  1) block-local inclusive scan of up_phase; quArr[t] = {S+frac(off), up}
//  2) wave-per-t: 32 lanes x 4 harmonics. amp rows streamed via an 8-deep
//     per-wave async global->LDS pipeline (ASYNCcnt), consumed with
//     ds_read_b128. v_sin_f32 on v_fract of phase, DPP wave reduction.
// LDS: quArr 8K + abuf 32K ~= 40KB/block.
// ---------------------------------------------------------------------------
__global__ __launch_bounds__(256) void k_main(const float* __restrict__ up,
                                              const float* __restrict__ amp,
                                              const float* __restrict__ init,
                                              const float* __restrict__ off,
                                              const float* __restrict__ chunk_prefix,
                                              float* __restrict__ out) {
    __shared__ float2 quArr[CH];  // {cumsum + frac(offset), up_phase} per t
    __shared__ float wsum[8];

    const int tid  = threadIdx.x;
    const int lane = tid & 31;
    const int wid  = tid >> 5;
    const int b    = blockIdx.x / CPB;
    const int t0   = (blockIdx.x % CPB) * CH;
    const long rowBase = (long)b * TT + t0;

    // ---- stage 1: scan 1024 up_phase values ----
    const int i = tid * 4;
    const float4 u = *(const float4*)(up + rowBase + i);
    const float4 o = *(const float4*)(off + rowBase + i);

    const float p0 = u.x, p1 = p0 + u.y, p2 = p1 + u.z, p3 = p2 + u.w;
    float v = p3;                                  // wave inclusive scan of thread totals
#pragma unroll
    for (int d = 1; d < 32; d <<= 1) {
        float n = __shfl_up(v, (unsigned)d, 32);
        if (lane >= d) v += n;
    }
    if (lane == 31) wsum[wid] = v;
    __syncthreads();
    float woff = 0.f;
    for (int w = 0; w < wid; ++w) woff += wsum[w];
    const float basePfx = chunk_prefix[blockIdx.x] + woff + (v - p3);  // exclusive for this thread
    quArr[i + 0] = make_float2(basePfx + p0 + fracf(o.x), u.x);
    quArr[i + 1] = make_float2(basePfx + p1 + fracf(o.y), u.y);
    quArr[i + 2] = make_float2(basePfx + p2 + fracf(o.z), u.z);
    quArr[i + 3] = make_float2(basePfx + p3 + fracf(o.w), u.w);
    __syncthreads();

    // ---- stage 2: harmonic sum, one wave per t ----
    const float k0 = (float)(lane * 4 + 1);                 // harmonics k..k+3
    const float4 in4 = *(const float4*)(init + b * HH + lane * 4);
    const float4* __restrict__ amp4 = (const float4*)amp;   // float4 view of [B*T, H]
    const int nIter = CH / 8;                               // 128 rows per wave

    // Inner body: amp row (float4 per lane) + per-t scalars -> out[row]
    auto harmRow = [&](float4 a, int t, long row) {
        const float2 qu = quArr[t];                         // one broadcast ds_read_b64
        const float Q = qu.x;
        const float U = qu.y;
        float acc = 0.f;
#pragma unroll
        for (int j = 0; j < 4; ++j) {
            const float kk = k0 + (float)j;
            const float av0 = (j == 0) ? a.x : (j == 1) ? a.y : (j == 2) ? a.z : a.w;
            const float ii  = (j == 0) ? in4.x : (j == 1) ? in4.y : (j == 2) ? in4.z : in4.w;
            const float av = (U * kk < 0.5f) ? av0 : 0.0f;  // alias mask (harm >= 0.5 -> 0)
            const float ph = fmaf(Q, kk, ii);               // phase in revolutions
            acc = fmaf(__builtin_amdgcn_sinf(fracf(ph)), av, acc);  // v_sin_f32: sin(2*pi*frac)
        }
        acc = wave32_sum(acc);
        if (lane == 0) out[row] = acc;
    };

#if defined(HAVE_ASYNC_LDS)
    __shared__ float4 abuf[8][PIPE][32];                    // 32 KB: per-wave ring buffer

    const float4* __restrict__ gp = amp4 + (rowBase + wid) * (HH / 4) + lane;

    auto issueRow = [&](int n) {
        __builtin_amdgcn_global_load_async_to_lds_b128(
            (v4i_global)(gp + (long)n * (8 * HH / 4)),
            (v4i_lds)(&abuf[wid][n & (PIPE - 1)][lane]), 0, 0);
    };
    auto doRow = [&](int n) {
        const int t = wid + n * 8;
        harmRow(abuf[wid][n & (PIPE - 1)][lane], t, rowBase + t);
    };

    // prologue: fill the pipeline
#pragma unroll
    for (int n = 0; n < PIPE; ++n) issueRow(n);
    // steady state: wait for oldest (loads land in order), compute, refill.
    // Unrolled by PIPE so ring-slot LDS addresses are compile-time constants.
#pragma unroll PIPE
    for (int n = 0; n < nIter - PIPE; ++n) {
        __builtin_amdgcn_s_wait_asynccnt(PIPE - 1);
        doRow(n);
        issueRow(n + PIPE);
    }
    // epilogue: drain
    __builtin_amdgcn_s_wait_asynccnt(0);
#pragma unroll
    for (int n = nIter - PIPE; n < nIter; ++n) doRow(n);
#else
    // Fallback: direct global b128 loads, unrolled for memory-level parallelism.
#pragma unroll 4
    for (int n = 0; n < nIter; ++n) {
        const int t = wid + n * 8;
        const long row = rowBase + t;
        const float4 a = amp4[row * (HH / 4) + lane];
        if (t + 64 < CH)
            __builtin_prefetch(amp + (row + 64) * (long)HH + (long)lane * 4, 0, 1);
        harmRow(a, t, row);
    }
#endif
}

// ---------------------------------------------------------------------------
// Launch
// ---------------------------------------------------------------------------
extern "C" void kernel_launch(void* const* d_in, const int* in_sizes, int n_in,
                              void* d_out, int out_size, void* d_ws, size_t ws_size,
                              hipStream_t stream) {
    const float* up   = (const float*)d_in[0];   // [B,T]
    const float* amp  = (const float*)d_in[1];   // [B,T,H]
    const float* init = (const float*)d_in[2];   // [B,H]
    const float* off  = (const float*)d_in[3];   // [B,T]
    // d_in[4] = hop_length (==1), unused.
    float* out = (float*)d_out;

    float* chunk_sums   = (float*)d_ws;          // NCHUNK floats
    float* chunk_prefix = chunk_sums + NCHUNK;   // NCHUNK floats

    k_chunk_sums <<<NCHUNK, 256, 0, stream>>>(up, chunk_sums);
    k_scan_chunks<<<1, NCHUNK, 0, stream>>>(chunk_sums, chunk_prefix);
    k_main       <<<NCHUNK, 256, 0, stream>>>(up, amp, init, off, chunk_prefix, out);
}